// RankNet_25735444037693
// MI455X (gfx1250) — compile-verified
//
#include <hip/hip_runtime.h>
#include <hip/hip_bf16.h>
#include <math.h>

// RankNet loss, but only logs[:5,0] is ever read by the reference -> we only
// need scores[0..4]. Entire job ~70 KFLOPs: single wave32, FP32 WMMA
// (v_wmma_f32_16x16x4_f32) for the 16x50x136 GEMM slice, LDS reduction,
// scalar tail on lane 0. Launch-latency bound; one workgroup is optimal.

typedef __attribute__((ext_vector_type(2))) float v2f;
typedef __attribute__((ext_vector_type(8))) float v8f;

#define DIM 136
#define HID 50

__global__ __launch_bounds__(32)
void ranknet_loss_kernel(const float* __restrict__ data,
                         const float* __restrict__ W1,
                         const float* __restrict__ bias1,
                         const float* __restrict__ W2,
                         const float* __restrict__ bias2,
                         float* __restrict__ out) {
    const int lane = threadIdx.x;        // 0..31, one wave
    const int row  = lane & 15;          // N-index within tile / M-row for A
    const int hi   = lane >> 4;          // which K-pair this lane holds
    const int kkb  = hi * 2;             // K offset within a K=4 step

    // Hidden-unit indices for the 4 N-tiles handled by this lane's column.
    const int h0 = row;                  // 0..15   (< 50 always)
    const int h1 = 16 + row;             // 16..31  (< 50 always)
    const int h2 = 32 + row;             // 32..47  (< 50 always)
    const int h3 = 48 + row;             // 48..63  (valid only if < 50)
    const int h3c   = (h3 < HID) ? h3 : 0;      // clamped (avoid OOB loads)
    const float m3  = (h3 < HID) ? 1.0f : 0.0f; // mask (keeps EXEC uniform)

    const float* arow  = data + row * DIM;      // rows 0..15 all valid (N=8192)
    const float* wrow0 = W1 + h0  * DIM;
    const float* wrow1 = W1 + h1  * DIM;
    const float* wrow2 = W1 + h2  * DIM;
    const float* wrow3 = W1 + h3c * DIM;

    v8f acc0 = {}, acc1 = {}, acc2 = {}, acc3 = {};

    // 136 = 34 K-steps of 4. D = A(16x4) * B(4x16) + C, all f32.
    for (int k = 0; k < DIM; k += 4) {
        v2f a;
        a.x = arow[k + kkb];
        a.y = arow[k + kkb + 1];

        v2f b0, b1, b2, b3;
        b0.x = wrow0[k + kkb];      b0.y = wrow0[k + kkb + 1];
        b1.x = wrow1[k + kkb];      b1.y = wrow1[k + kkb + 1];
        b2.x = wrow2[k + kkb];      b2.y = wrow2[k + kkb + 1];
        b3.x = wrow3[k + kkb] * m3; b3.y = wrow3[k + kkb + 1] * m3;

        acc0 = __builtin_amdgcn_wmma_f32_16x16x4_f32(false, a, false, b0, (short)0, acc0, false, false);
        acc1 = __builtin_amdgcn_wmma_f32_16x16x4_f32(false, a, false, b1, (short)0, acc1, false, false);
        acc2 = __builtin_amdgcn_wmma_f32_16x16x4_f32(false, a, false, b2, (short)0, acc2, false, false);
        acc3 = __builtin_amdgcn_wmma_f32_16x16x4_f32(false, a, false, b3, (short)0, acc3, false, false);
    }

    // Epilogue: h = tanh(pre + b1); partial score contribution = h * W2[h].
    // C/D layout: VGPR v, lanes 0-15 -> M=v, lanes 16-31 -> M=v+8.
    const float bb0 = bias1[h0],  ww0 = W2[h0];
    const float bb1 = bias1[h1],  ww1 = W2[h1];
    const float bb2 = bias1[h2],  ww2 = W2[h2];
    const float bb3 = bias1[h3c] * m3, ww3 = W2[h3c] * m3;

    // Only m = 0..4 are needed -> slots v=0..4 on lanes 0..15 (hi==0).
    __shared__ float red[16][5];
    float p[5];
#pragma unroll
    for (int v = 0; v < 5; ++v) {
        float s = 0.0f;
        s += tanhf(acc0[v] + bb0) * ww0;
        s += tanhf(acc1[v] + bb1) * ww1;
        s += tanhf(acc2[v] + bb2) * ww2;
        s += tanhf(acc3[v] + bb3) * ww3;   // masked to 0 when h3 >= 50
        p[v] = s;
    }
    if (hi == 0) {
#pragma unroll
        for (int v = 0; v < 5; ++v) red[row][v] = p[v];
    }
    __syncthreads();

    if (lane == 0) {
        float scores[5];
#pragma unroll
        for (int i = 0; i < 5; ++i) {
            float acc = 0.0f;
            for (int l = 0; l < 16; ++l) acc += red[l][i];
            scores[i] = acc + bias2[0];
        }
        const float y = scores[0];
        float total = 0.0f;
#pragma unroll
        for (int i = 0; i < 5; ++i) {
            const float x = scores[i];
            // Match the reference's f32 expression order exactly.
            float s = x * x * x - 3.0f * (x * x) * y + 3.0f * x * (y * y) - y * y * y;
            float sgn = (s > 0.0f) ? 1.0f : ((s < 0.0f) ? -1.0f : 0.0f);
            float dist = powf(fabsf(s + 1e-6f), 0.33f) * sgn;
            total += logf(1.0f + expf(-dist));  // gamma = 1
        }
        out[0] = total * (1.0f / (8192.0f * 8192.0f));  // exact 2^-26
    }
}

extern "C" void kernel_launch(void* const* d_in, const int* in_sizes, int n_in,
                              void* d_out, int out_size, void* d_ws, size_t ws_size,
                              hipStream_t stream) {
    (void)in_sizes; (void)n_in; (void)d_ws; (void)ws_size; (void)out_size;
    const float* data  = (const float*)d_in[0];  // (8192, 136)
    // d_in[1] = ones, d_in[2] = zeros : unused by the reference
    const float* W1    = (const float*)d_in[3];  // (50, 136)
    const float* b1    = (const float*)d_in[4];  // (50,)
    const float* W2    = (const float*)d_in[5];  // (1, 50)
    const float* b2    = (const float*)d_in[6];  // (1,)
    float* out = (float*)d_out;

    ranknet_loss_kernel<<<1, 32, 0, stream>>>(data, W1, b1, W2, b2, out);
}